// RNN_1675037245608
// MI455X (gfx1250) — compile-verified
//
#include <hip/hip_runtime.h>
#include <math.h>

// Problem constants (match reference)
#define TT   8192
#define HH   32
#define DD   64
#define SS   2048   // H*D
#define INF_ 2048
#define OUTF 2048
#define PROJ_W 4096 // 2*S
#define EPSV 1e-6f

typedef float v2f __attribute__((ext_vector_type(2)));
typedef float v8f __attribute__((ext_vector_type(8)));
typedef int   v4i __attribute__((ext_vector_type(4)));

// ---- gfx1250 async-to-LDS availability probes ------------------------------
#if defined(__has_builtin)
#if __has_builtin(__builtin_amdgcn_global_load_async_to_lds_b128)
#define HAVE_ASYNC_LDS 1
#endif
#if __has_builtin(__builtin_amdgcn_s_wait_asynccnt)
#define HAVE_ASYNC_WAIT 1
#endif
#endif
#ifndef HAVE_ASYNC_LDS
#define HAVE_ASYNC_LDS 0
#endif
#ifndef HAVE_ASYNC_WAIT
#define HAVE_ASYNC_WAIT 0
#endif

#if HAVE_ASYNC_LDS
// From the clang diagnostic: src param is int4 in addrspace(1) ("__device__"),
// dst is the LDS-side counterpart in addrspace(3).
typedef __attribute__((address_space(1))) v4i gv4i;
typedef __attribute__((address_space(3))) v4i lv4i;
#endif

// XOR-granule swizzle: element (row, k) lives at granule (k/4) ^ ((row>>2)&3).
// Keeps every 16B transfer aligned and makes the 16-consecutive-row fragment
// reads (stride 16 floats = bank wrap every 4 rows) conflict-free.
__device__ __forceinline__ int sw_off(int row, int g) {
  return (row << 4) + (((g ^ (row >> 2)) & 3) << 2);
}

// -----------------------------------------------------------------------------
// C[M x N] = A[M x K] * W[N x K]^T   (fp32, V_WMMA_F32_16X16X4_F32)
// 256 threads (8 waves), 128x64 output tile. K in chunks of 16, double-buffered
// LDS fed by GLOBAL_LOAD_ASYNC_TO_LDS_B128 (s_wait_asynccnt), swizzled layout.
// Each wave owns a 32x32 register tile = 2x2 WMMA subtiles.
// -----------------------------------------------------------------------------
__global__ __launch_bounds__(256) void gemm_xwT(const float* __restrict__ A,
                                                const float* __restrict__ W,
                                                float* __restrict__ C,
                                                int K, int lda, int ldc) {
  __shared__ float As[2][128 * 16];  // 2 x 8KB
  __shared__ float Bs[2][64 * 16];   // 2 x 4KB

  const int tileM = blockIdx.y * 128;
  const int tileN = blockIdx.x * 64;
  const int tid   = threadIdx.x;
  const int wid   = tid >> 5;
  const int lane  = tid & 31;
  const int m     = lane & 15;       // subtile row (A) / subtile col (B)
  const int half  = lane >> 4;       // K-pair select within WMMA fragment
  const int h2    = half * 2;

  const int mstrip0 = (wid >> 1) * 32;     // wave's two M strips: +0, +16
  const int nbase   = (wid & 1) * 32;      // wave's two N subtiles: +0, +16

  // ---- cooperative loader mapping (3 x 16B granules per thread per chunk) --
  const int arow = tid >> 1;               // 0..127
  const int ag0  = (tid & 1) * 2;          // granules ag0, ag0+1
  const int brow = tid >> 2;               // 0..63
  const int bg   = tid & 3;                // one granule

  const float* aG = A + (size_t)(tileM + arow) * lda + ag0 * 4;
  const float* bG = W + (size_t)(tileN + brow) * K   + bg  * 4;

  const int aL0 = sw_off(arow, ag0);
  const int aL1 = sw_off(arow, ag0 + 1);
  const int bL  = sw_off(brow, bg);

  auto load_chunk = [&](int buf, int k0) {
#if HAVE_ASYNC_LDS
    __builtin_amdgcn_global_load_async_to_lds_b128(
        (gv4i*)(void*)(aG + k0),     (lv4i*)(void*)&As[buf][aL0], 0, 0);
    __builtin_amdgcn_global_load_async_to_lds_b128(
        (gv4i*)(void*)(aG + k0 + 4), (lv4i*)(void*)&As[buf][aL1], 0, 0);
    __builtin_amdgcn_global_load_async_to_lds_b128(
        (gv4i*)(void*)(bG + k0),     (lv4i*)(void*)&Bs[buf][bL],  0, 0);
#else
    const float4 a0 = *(const float4*)(aG + k0);
    const float4 a1 = *(const float4*)(aG + k0 + 4);
    const float4 b0 = *(const float4*)(bG + k0);
    *(float4*)&As[buf][aL0] = a0;
    *(float4*)&As[buf][aL1] = a1;
    *(float4*)&Bs[buf][bL]  = b0;
    if (k0 + 16 < K) {
      __builtin_prefetch(aG + k0 + 16, 0, 1);
      __builtin_prefetch(bG + k0 + 16, 0, 1);
    }
#endif
  };

  v8f acc00 = {0.f,0.f,0.f,0.f,0.f,0.f,0.f,0.f};
  v8f acc01 = acc00, acc10 = acc00, acc11 = acc00;

  const int r0 = mstrip0 + m;        // A rows for the two M strips
  const int r1 = mstrip0 + 16 + m;
  const int n0 = nbase + m;          // B rows (= output cols) for N subtiles
  const int n1 = nbase + 16 + m;

  load_chunk(0, 0);

  for (int k0 = 0; k0 < K; k0 += 16) {
    const int buf = (k0 >> 4) & 1;
    const bool more = (k0 + 16) < K;
    if (more) load_chunk(buf ^ 1, k0 + 16);   // overlap next chunk
#if HAVE_ASYNC_LDS
#if HAVE_ASYNC_WAIT
    if (more) __builtin_amdgcn_s_wait_asynccnt(3);  // current chunk resident
    else      __builtin_amdgcn_s_wait_asynccnt(0);
#else
    asm volatile("s_wait_asynccnt 0x0" ::: "memory");
#endif
#endif
    __syncthreads();

#pragma unroll
    for (int ks = 0; ks < 4; ++ks) {
      const v2f a0 = *(const v2f*)&As[buf][sw_off(r0, ks) + h2];
      const v2f a1 = *(const v2f*)&As[buf][sw_off(r1, ks) + h2];
      const v2f b0 = *(const v2f*)&Bs[buf][sw_off(n0, ks) + h2];
      const v2f b1 = *(const v2f*)&Bs[buf][sw_off(n1, ks) + h2];
      acc00 = __builtin_amdgcn_wmma_f32_16x16x4_f32(false, a0, false, b0,
                                                    (short)0, acc00, false, false);
      acc01 = __builtin_amdgcn_wmma_f32_16x16x4_f32(false, a0, false, b1,
                                                    (short)0, acc01, false, false);
      acc10 = __builtin_amdgcn_wmma_f32_16x16x4_f32(false, a1, false, b0,
                                                    (short)0, acc10, false, false);
      acc11 = __builtin_amdgcn_wmma_f32_16x16x4_f32(false, a1, false, b1,
                                                    (short)0, acc11, false, false);
    }
    __syncthreads();  // all reads of buf done before it is refilled
  }

  // C/D layout: VGPR r -> M = r + half*8, N = lane&15
  float* cbase = C + (size_t)(tileM + mstrip0 + half * 8) * ldc + tileN + nbase + m;
#pragma unroll
  for (int r = 0; r < 8; ++r) {
    cbase[(size_t)r * ldc]              = acc00[r];
    cbase[(size_t)r * ldc + 16]         = acc01[r];
    cbase[(size_t)(r + 16) * ldc]       = acc10[r];
    cbase[(size_t)(r + 16) * ldc + 16]  = acc11[r];
  }
}

// -----------------------------------------------------------------------------
// Per-(sequence, head) linear recurrence:  y_t = W_h @ y_{t-1} + xi_t
// xi lives in proj[:, :2048] viewed as [T,H,D]; y written back in place.
// 64 threads: thread i owns output row i (W row in registers, state via LDS).
// -----------------------------------------------------------------------------
__global__ __launch_bounds__(64) void rnn_scan(float* __restrict__ proj,
                                               const float* __restrict__ Wst,
                                               const int* __restrict__ cu) {
  const int h = blockIdx.x;     // head
  const int s = blockIdx.y;     // sequence
  const int i = threadIdx.x;    // output element 0..63
  const int t0 = cu[s];
  const int t1 = cu[s + 1];

  float4 wr[16];
  const float4* wrow = (const float4*)(Wst + ((size_t)h * DD + i) * DD);
#pragma unroll
  for (int q = 0; q < 16; ++q) wr[q] = wrow[q];

  __shared__ float st[DD];
  st[i] = 0.f;
  __syncthreads();

  float* base = proj + (size_t)h * DD + i;
  for (int t = t0; t < t1; ++t) {
    float acc = base[(size_t)t * PROJ_W];
#pragma unroll
    for (int q = 0; q < 16; ++q) {
      acc = fmaf(wr[q].x, st[q * 4 + 0], acc);
      acc = fmaf(wr[q].y, st[q * 4 + 1], acc);
      acc = fmaf(wr[q].z, st[q * 4 + 2], acc);
      acc = fmaf(wr[q].w, st[q * 4 + 3], acc);
    }
    __syncthreads();          // all reads of st done
    st[i] = acc;
    base[(size_t)t * PROJ_W] = acc;
    __syncthreads();          // st updated before next step
  }
}

// -----------------------------------------------------------------------------
// v = y * silu(g);  y_norm = v * rsqrt(mean(v^2) + eps) * norm_weight
// One block per token; result overwrites proj[:, :2048] (feeds GEMM-2).
// -----------------------------------------------------------------------------
__global__ __launch_bounds__(256) void gate_rmsnorm(float* __restrict__ proj,
                                                    const float* __restrict__ nw) {
  const int t = blockIdx.x;
  float* yrow = proj + (size_t)t * PROJ_W;       // y in [:, :2048]
  const float* grow = yrow + SS;                 // g in [:, 2048:]

  __shared__ float red[256];
  float v[8];
  float ss = 0.f;
#pragma unroll
  for (int p = 0; p < 8; ++p) {
    const int j = threadIdx.x + p * 256;
    const float g  = grow[j];
    const float sg = g / (1.f + __expf(-g));     // silu
    const float val = yrow[j] * sg;
    v[p] = val;
    ss += val * val;
  }
  red[threadIdx.x] = ss;
  __syncthreads();
#pragma unroll
  for (int off = 128; off > 0; off >>= 1) {
    if (threadIdx.x < off) red[threadIdx.x] += red[threadIdx.x + off];
    __syncthreads();
  }
  const float scale = rsqrtf(red[0] * (1.0f / (float)SS) + EPSV);
#pragma unroll
  for (int p = 0; p < 8; ++p) {
    const int j = threadIdx.x + p * 256;
    yrow[j] = v[p] * scale * nw[j];
  }
}

// -----------------------------------------------------------------------------
extern "C" void kernel_launch(void* const* d_in, const int* in_sizes, int n_in,
                              void* d_out, int out_size, void* d_ws, size_t ws_size,
                              hipStream_t stream) {
  const float* x     = (const float*)d_in[0];  // [8192, 2048]
  const float* w_in  = (const float*)d_in[1];  // [4096, 2048]
  const float* sw    = (const float*)d_in[2];  // [32, 64, 64]
  const float* nw    = (const float*)d_in[3];  // [2048]
  const float* w_out = (const float*)d_in[4];  // [2048, 2048]
  const int*   cu    = (const int*)d_in[5];    // [nseq+1]
  float* out  = (float*)d_out;                 // [8192, 2048]
  float* proj = (float*)d_ws;                  // [8192, 4096] scratch

  const int nseq = in_sizes[5] - 1;

  // 1) proj = x @ w_in.T   (M=8192, N=4096, K=2048)
  gemm_xwT<<<dim3(PROJ_W / 64, TT / 128), 256, 0, stream>>>(
      x, w_in, proj, INF_, INF_, PROJ_W);

  // 2) in-place linear RNN scan over xi = proj[:, :2048]
  rnn_scan<<<dim3(HH, nseq), 64, 0, stream>>>(proj, sw, cu);

  // 3) gate + rmsnorm, in place into proj[:, :2048]
  gate_rmsnorm<<<TT, 256, 0, stream>>>(proj, nw);

  // 4) out = y_norm @ w_out.T  (M=8192, N=2048, K=2048, lda=4096)
  gemm_xwT<<<dim3(OUTF / 64, TT / 128), 256, 0, stream>>>(
      proj, w_out, out, SS, PROJ_W, OUTF);
}